// WeightedGCN_40441412059454
// MI455X (gfx1250) — compile-verified
//
#include <hip/hip_runtime.h>

#define TT 3
#define F 64
#define EPS 1e-5f

typedef __attribute__((ext_vector_type(2))) float v2f;
typedef __attribute__((ext_vector_type(8))) float v8f;

// ---------------------------------------------------------------- zero
__global__ __launch_bounds__(256) void zero_kernel(float4* p, int n4) {
    int i = blockIdx.x * blockDim.x + threadIdx.x;
    if (i < n4) p[i] = make_float4(0.f, 0.f, 0.f, 0.f);
}

// ---------------------------------------------------------------- edge scatter
// One thread per (edge, t, quad-of-4-channels): 16 threads cover the 64-float
// row -> fully coalesced float4 gather, 4 native f32 atomics to the dst row.
__global__ __launch_bounds__(256) void scatter_kernel(
    const float* __restrict__ h, const float* __restrict__ ew,
    const int* __restrict__ src, const int* __restrict__ dst,
    float* __restrict__ agg, int E)
{
    int idx = blockIdx.x * blockDim.x + threadIdx.x;
    int total = E * TT * 16;
    if (idx >= total) return;
    int q  = idx & 15;
    int et = idx >> 4;
    int e  = et / TT;
    int t  = et - e * TT;
    int s = src[e];
    int d = dst[e];
    float w = ew[t * E + e];
    float4 v = ((const float4*)(h + (s * TT + t) * F))[q];
    float* ap = agg + (d * TT + t) * F + q * 4;
    unsafeAtomicAdd(ap + 0, v.x * w);
    unsafeAtomicAdd(ap + 1, v.y * w);
    unsafeAtomicAdd(ap + 2, v.z * w);
    unsafeAtomicAdd(ap + 3, v.w * w);
}

// ---------------------------------------------------------------- GEMM (WMMA fp32) + BN stats
// out[m][n] = sum_k A[m][k] * W[n][k] + bias[n]
// One wave32 per block computes a 16x64 tile via V_WMMA_F32_16X16X4_F32.
// A lane layout (16x4):  lanes 0-15: row=lane, K={0,1}; lanes 16-31: row=lane-16, K={2,3}
// B lane layout (4x16):  lanes 0-15: col=lane, K={0,1}; lanes 16-31: col=lane-16, K={2,3}
//   B[k][n] = W[n*64+k] -> float2 load from W[col*64 + k] (contiguous)
// C/D layout: vgpr r -> row = r + 8*(lane>=16), col = lane&15
__global__ __launch_bounds__(32) void gemm_bn_kernel(
    const float* __restrict__ A, const float* __restrict__ W,
    const float* __restrict__ bias, float* __restrict__ Hpre,
    float* __restrict__ stats)
{
    int m0   = blockIdx.x * 16;
    int lane = threadIdx.x;
    int half = lane >> 4;   // 0 or 1
    int l    = lane & 15;

    v8f acc0 = {}, acc1 = {}, acc2 = {}, acc3 = {};
    const float* arow = A + (m0 + l) * F + 2 * half;

    for (int kk = 0; kk < 16; ++kk) {
        int k = kk * 4;
        v2f a = *(const v2f*)(arow + k);
        v2f b0 = *(const v2f*)(W + (l     ) * F + k + 2 * half);
        v2f b1 = *(const v2f*)(W + (l + 16) * F + k + 2 * half);
        v2f b2 = *(const v2f*)(W + (l + 32) * F + k + 2 * half);
        v2f b3 = *(const v2f*)(W + (l + 48) * F + k + 2 * half);
        acc0 = __builtin_amdgcn_wmma_f32_16x16x4_f32(false, a, false, b0, (short)0, acc0, false, false);
        acc1 = __builtin_amdgcn_wmma_f32_16x16x4_f32(false, a, false, b1, (short)0, acc1, false, false);
        acc2 = __builtin_amdgcn_wmma_f32_16x16x4_f32(false, a, false, b2, (short)0, acc2, false, false);
        acc3 = __builtin_amdgcn_wmma_f32_16x16x4_f32(false, a, false, b3, (short)0, acc3, false, false);
    }

    v8f accs[4] = {acc0, acc1, acc2, acc3};
    #pragma unroll
    for (int g = 0; g < 4; ++g) {
        int col = 16 * g + l;
        float bc = bias[col];
        float s = 0.f, ss = 0.f;
        #pragma unroll
        for (int r = 0; r < 8; ++r) {
            float v = accs[g][r] + bc;
            int row = m0 + r + 8 * half;
            Hpre[row * F + col] = v;
            s  += v;
            ss += v * v;
        }
        unsafeAtomicAdd(&stats[col],      s);
        unsafeAtomicAdd(&stats[64 + col], ss);
    }
}

// ---------------------------------------------------------------- BN finalize (64 threads)
__global__ void bn_finalize_kernel(float* stats, const float* __restrict__ gamma,
                                   const float* __restrict__ beta, float inv_count)
{
    int c = threadIdx.x;
    if (c >= F) return;
    float mean  = stats[c] * inv_count;
    float var   = stats[64 + c] * inv_count - mean * mean;
    float scale = gamma[c] * rsqrtf(var + EPS);
    float shift = beta[c] - mean * scale;
    stats[c]      = scale;
    stats[64 + c] = shift;
}

// ---------------------------------------------------------------- normalize + ReLU
__global__ __launch_bounds__(256) void bn_relu_kernel(
    const float4* __restrict__ hp, const float* __restrict__ sc,
    float4* out, int n4)
{
    int i = blockIdx.x * blockDim.x + threadIdx.x;
    if (i >= n4) return;
    float4 v = hp[i];
    int c = (i & 15) * 4;   // (i*4) % 64
    float4 o;
    o.x = fmaxf(0.f, v.x * sc[c + 0] + sc[64 + c + 0]);
    o.y = fmaxf(0.f, v.y * sc[c + 1] + sc[64 + c + 1]);
    o.z = fmaxf(0.f, v.z * sc[c + 2] + sc[64 + c + 2]);
    o.w = fmaxf(0.f, v.w * sc[c + 3] + sc[64 + c + 3]);
    out[i] = o;
}

// ---------------------------------------------------------------- launch
extern "C" void kernel_launch(void* const* d_in, const int* in_sizes, int n_in,
                              void* d_out, int out_size, void* d_ws, size_t ws_size,
                              hipStream_t stream) {
    const float* nf  = (const float*)d_in[0];
    const float* ew  = (const float*)d_in[1];
    const int*   src = (const int*)d_in[2];
    const int*   dst = (const int*)d_in[3];
    const float* W[3]  = {(const float*)d_in[4],  (const float*)d_in[8],  (const float*)d_in[12]};
    const float* b[3]  = {(const float*)d_in[5],  (const float*)d_in[9],  (const float*)d_in[13]};
    const float* gm[3] = {(const float*)d_in[6],  (const float*)d_in[10], (const float*)d_in[14]};
    const float* bt[3] = {(const float*)d_in[7],  (const float*)d_in[11], (const float*)d_in[15]};

    int E = in_sizes[2];
    int M = in_sizes[0] / F;          // N * T rows

    float* bufA  = (float*)d_ws;                  // aggregation buffer
    float* bufB  = bufA + (size_t)M * F;          // pre-BN / activation buffer
    float* stats = bufB + (size_t)M * F;          // 128 floats: sum/sumsq -> scale/shift

    int n4 = M * F / 4;
    int gz = (n4 + 255) / 256;
    int gs = (E * TT * 16 + 255) / 256;
    int gg = M / 16;                  // 150000/16 = 9375, exact

    const float* hin = nf;
    for (int l = 0; l < 3; ++l) {
        zero_kernel<<<gz, 256, 0, stream>>>((float4*)bufA, n4);
        zero_kernel<<<1, 32, 0, stream>>>((float4*)stats, 32);
        scatter_kernel<<<gs, 256, 0, stream>>>(hin, ew, src, dst, bufA, E);
        gemm_bn_kernel<<<gg, 32, 0, stream>>>(bufA, W[l], b[l], bufB, stats);
        bn_finalize_kernel<<<1, 64, 0, stream>>>(stats, gm[l], bt[l], 1.0f / (float)M);
        float* hout = (l == 2) ? (float*)d_out : bufB;
        bn_relu_kernel<<<gz, 256, 0, stream>>>((const float4*)bufB, stats, (float4*)hout, n4);
        hin = hout;
    }
}